// SPDNet_58394375356769
// MI455X (gfx1250) — compile-verified
//
#include <hip/hip_runtime.h>
#include <math.h>
#include <stdint.h>

// ---------------------------------------------------------------------------
// SPDNet fused kernel for gfx1250 (MI455X, wave32, WMMA, async LDS DMA).
//   cov+BiMap folded:  Y = (1/T) (W^T x)(W^T x)^T - zbar zbar^T
//   ReEig+LogEig folded: one Jacobi eigensolve, f(s) = log(max(s, 1e-4))
//   GEMMs:  V_WMMA_F32_16X16X4_F32 (fp32 precision matches reference)
//   x tiles: GLOBAL_LOAD_ASYNC_TO_LDS_B128, double buffered, s_wait_asynccnt
// One wave (32 lanes) per sample; 4096 blocks.
// ---------------------------------------------------------------------------

typedef float v2f __attribute__((ext_vector_type(2)));
typedef float v8f __attribute__((ext_vector_type(8)));

#define C_ 64
#define T_ 256
#define D_ 32
#define O_ 4
#define THRESH 1e-4f

// LDS layout (float offsets). Regions are reused across phases.
#define WS_OFF 0      // W staging [64][36]      ; later F = U log(L) U^T [32][34]
#define X0_OFF 2304   // x chunk ping [64][36]   ; later A (the SPD Y)     [32][34]
#define X1_OFF 4608   // x chunk pong [64][36]
#define ZS_OFF 6912   // z chunk [32][34]
#define US_OFF 8000   // U eigvecs [32][34]
#define SMEM_FLOATS 9088

__device__ __forceinline__ v8f wmma4(v2f a, v2f b, v8f c) {
  // D = A(16x4 f32) * B(4x16 f32) + C(16x16 f32)
  return __builtin_amdgcn_wmma_f32_16x16x4_f32(
      /*neg_a=*/false, a, /*neg_b=*/false, b,
      /*c_mod=*/(short)0, c, /*reuse_a=*/false, /*reuse_b=*/false);
}

// Issue 16 async B128 copies: x chunk [64 rows][32 cols] -> LDS (stride 36).
// Each instruction moves 16B/lane = 512B; data lands without touching VGPRs.
__device__ __forceinline__ void async_load_x_chunk(const float* xg, uint32_t lds_byte_base,
                                                   int ct, int lid) {
#pragma unroll
  for (int i = 0; i < 16; ++i) {
    int f = i * 32 + lid;
    int row = f >> 3, c4 = f & 7;            // 8 float4 per 32-float row
    const float* gp = xg + row * T_ + ct * 32 + c4 * 4;
    uint32_t lp = lds_byte_base + (uint32_t)(row * 36 + c4 * 4) * 4u;  // 16B aligned
    asm volatile("global_load_async_to_lds_b128 %0, %1, off"
                 :: "v"(lp), "v"(gp)
                 : "memory");
  }
}

__global__ __launch_bounds__(32)
void spdnet_fused(const float* __restrict__ x, const float* __restrict__ W,
                  const float* __restrict__ Wc, const float* __restrict__ bc,
                  float* __restrict__ out) {
  __shared__ float smem[SMEM_FLOATS];
  __shared__ float cs_s[32];    // (c,s) per rotation pair
  __shared__ float lam_s[32];   // zsum during finalize; log-eigenvalues later
  __shared__ float logit_s[O_];
  __shared__ int tp_s[16], bt_s[16];  // round-robin tournament slots

  const int lid = threadIdx.x;
  const int b   = blockIdx.x;
  const int hh  = lid >> 4;   // half-wave: selects K-pair for WMMA frags
  const int lr  = lid & 15;   // lane-in-half: M/N index inside 16x16 tile

  float* Wsm = &smem[WS_OFF];
  float* Zsm = &smem[ZS_OFF];
  float* Usm = &smem[US_OFF];
  float* Asm = &smem[X0_OFF];  // alias (post chunk loop)
  float* Fsm = &smem[WS_OFF];  // alias (post Jacobi)

  const float* xg = x + (size_t)b * (C_ * T_);
  // LDS aperture tag is in addr[63:32]; addr[31:0] is the LDS byte address.
  const uint32_t smem_base = (uint32_t)(uintptr_t)&smem[0];
  const uint32_t xb[2] = { smem_base + X0_OFF * 4u, smem_base + X1_OFF * 4u };

  // kick off chunk 0 DMA before anything else
  async_load_x_chunk(xg, xb[0], 0, lid);

  // ---- load W [64][32] row-major -> Wsm stride 36 (float4, coalesced) ----
  {
    const float4* Wg = (const float4*)W;
#pragma unroll
    for (int i = 0; i < 16; ++i) {
      int f = i * 32 + lid;
      int row = f >> 3, c4 = f & 7;
      float4 v = Wg[row * 8 + c4];
      *(float4*)&Wsm[row * 36 + c4 * 4] = v;
    }
  }

  const v8f vzero = {0.f, 0.f, 0.f, 0.f, 0.f, 0.f, 0.f, 0.f};
  v8f yacc[2][2];
  yacc[0][0] = vzero; yacc[0][1] = vzero;
  yacc[1][0] = vzero; yacc[1][1] = vzero;
  float zsum = 0.f;  // per-lane row-sum of z (lane = row d)

  // ================= one pass over x: 8 chunks of 32 time-columns =========
#pragma unroll 1
  for (int ct = 0; ct < T_ / 32; ++ct) {
    float* Xsm = (ct & 1) ? &smem[X1_OFF] : &smem[X0_OFF];
    if (ct < T_ / 32 - 1) {
      // prefetch next chunk into the other buffer, then wait so that only the
      // 16 just-issued copies may remain in flight => chunk ct has landed
      // (async loads complete in order).
      async_load_x_chunk(xg, xb[(ct + 1) & 1], ct + 1, lid);
      asm volatile("s_wait_asynccnt 16" ::: "memory");
    } else {
      asm volatile("s_wait_asynccnt 0" ::: "memory");
    }
    __syncthreads();

    // ---- z = W^T * xchunk : (32x64)(64x32) via 16x16x4 f32 WMMA ----------
    v8f zacc[2][2];
    zacc[0][0] = vzero; zacc[0][1] = vzero;
    zacc[1][0] = vzero; zacc[1][1] = vzero;
#pragma unroll 4
    for (int kk = 0; kk < 16; ++kk) {
      int k0 = kk * 4 + hh * 2;  // A frag: lanes 0-15 K0/K1, 16-31 K2/K3
      v2f a0, a1, b0, b1;
      a0.x = Wsm[k0 * 36 + lr];            a0.y = Wsm[(k0 + 1) * 36 + lr];
      a1.x = Wsm[k0 * 36 + 16 + lr];       a1.y = Wsm[(k0 + 1) * 36 + 16 + lr];
      b0.x = Xsm[k0 * 36 + lr];            b0.y = Xsm[(k0 + 1) * 36 + lr];
      b1.x = Xsm[k0 * 36 + 16 + lr];       b1.y = Xsm[(k0 + 1) * 36 + 16 + lr];
      zacc[0][0] = wmma4(a0, b0, zacc[0][0]);
      zacc[0][1] = wmma4(a0, b1, zacc[0][1]);
      zacc[1][0] = wmma4(a1, b0, zacc[1][0]);
      zacc[1][1] = wmma4(a1, b1, zacc[1][1]);
    }
    // spill z tiles to LDS (C/D layout: VGPR r -> M = r + 8*half, N = lr)
#pragma unroll
    for (int mi = 0; mi < 2; ++mi)
#pragma unroll
      for (int ni = 0; ni < 2; ++ni)
#pragma unroll
        for (int r = 0; r < 8; ++r)
          Zsm[(mi * 16 + r + 8 * hh) * 34 + ni * 16 + lr] = zacc[mi][ni][r];
    __syncthreads();

    // ---- Y += z z^T : A and B frags are identical reads (symmetric) ------
#pragma unroll 4
    for (int kk = 0; kk < 8; ++kk) {
      int k0 = kk * 4 + hh * 2;
      v2f f0, f1;
      f0.x = Zsm[lr * 34 + k0];            f0.y = Zsm[lr * 34 + k0 + 1];
      f1.x = Zsm[(16 + lr) * 34 + k0];     f1.y = Zsm[(16 + lr) * 34 + k0 + 1];
      yacc[0][0] = wmma4(f0, f0, yacc[0][0]);
      yacc[0][1] = wmma4(f0, f1, yacc[0][1]);
      yacc[1][0] = wmma4(f1, f0, yacc[1][0]);
      yacc[1][1] = wmma4(f1, f1, yacc[1][1]);
    }
    // accumulate z row sums (lane lid owns row lid)
#pragma unroll 8
    for (int n = 0; n < 32; ++n) zsum += Zsm[lid * 34 + n];
    __syncthreads();
  }

  // ================= finalize Y = yacc/T - zbar zbar^T =====================
  lam_s[lid] = zsum;
  __syncthreads();
  const float invT = 1.0f / (float)T_;
#pragma unroll
  for (int mi = 0; mi < 2; ++mi)
#pragma unroll
    for (int ni = 0; ni < 2; ++ni)
#pragma unroll
      for (int r = 0; r < 8; ++r) {
        int row = mi * 16 + r + 8 * hh;
        int col = ni * 16 + lr;
        float zr = lam_s[row] * invT;
        float zc = lam_s[col] * invT;
        Asm[row * 34 + col] = yacc[mi][ni][r] * invT - zr * zc;
      }
  // init U = I and tournament pairing
  for (int i = 0; i < 32; ++i) Usm[i * 34 + lid] = (i == lid) ? 1.f : 0.f;
  if (lid < 16) { tp_s[lid] = lid; bt_s[lid] = lid + 16; }
  __syncthreads();

  // ================= parallel-order Jacobi eigensolve ======================
  // 16 disjoint rotations per step; 31 steps/sweep cover all 496 pairs.
#pragma unroll 1
  for (int sweep = 0; sweep < 8; ++sweep) {
#pragma unroll 1
    for (int round = 0; round < 31; ++round) {
      if (lid < 16) {
        int p = tp_s[lid], q = bt_s[lid];
        float app = Asm[p * 34 + p], aqq = Asm[q * 34 + q], apq = Asm[p * 34 + q];
        float c = 1.f, s = 0.f;
        if (fabsf(apq) > 1e-12f) {
          float tau = (aqq - app) / (2.f * apq);
          float t = copysignf(1.f, tau) / (fabsf(tau) + sqrtf(1.f + tau * tau));
          c = rsqrtf(1.f + t * t);
          s = t * c;
        }
        cs_s[2 * lid] = c; cs_s[2 * lid + 1] = s;
      }
      __syncthreads();
      // row phase: G^T A   (lane = column j; pairs are disjoint)
#pragma unroll 4
      for (int k = 0; k < 16; ++k) {
        int p = tp_s[k], q = bt_s[k];
        float c = cs_s[2 * k], s = cs_s[2 * k + 1];
        float Ap = Asm[p * 34 + lid], Aq = Asm[q * 34 + lid];
        Asm[p * 34 + lid] = c * Ap - s * Aq;
        Asm[q * 34 + lid] = s * Ap + c * Aq;
      }
      __syncthreads();
      // col phase: (.) G  on A and U (lane = row j)
#pragma unroll 4
      for (int k = 0; k < 16; ++k) {
        int p = tp_s[k], q = bt_s[k];
        float c = cs_s[2 * k], s = cs_s[2 * k + 1];
        float Ap = Asm[lid * 34 + p], Aq = Asm[lid * 34 + q];
        Asm[lid * 34 + p] = c * Ap - s * Aq;
        Asm[lid * 34 + q] = s * Ap + c * Aq;
        float Up = Usm[lid * 34 + p], Uq = Usm[lid * 34 + q];
        Usm[lid * 34 + p] = c * Up - s * Uq;
        Usm[lid * 34 + q] = s * Up + c * Uq;
      }
      __syncthreads();
      // rotate tournament slots (t[0] fixed)
      if (lid == 0) {
        int t15 = tp_s[15];
        for (int k = 15; k >= 2; --k) tp_s[k] = tp_s[k - 1];
        tp_s[1] = bt_s[0];
        for (int k = 0; k < 15; ++k) bt_s[k] = bt_s[k + 1];
        bt_s[15] = t15;
      }
      __syncthreads();
    }
  }

  // ================= F = U log(clamp(L)) U^T ===============================
  float lam = logf(fmaxf(Asm[lid * 34 + lid], THRESH));
  __syncthreads();
  lam_s[lid] = lam;
  __syncthreads();
  float uj[32];
#pragma unroll
  for (int k = 0; k < 32; ++k) uj[k] = Usm[lid * 34 + k] * lam_s[k];
#pragma unroll 1
  for (int i = 0; i < 32; ++i) {
    float acc = 0.f;
#pragma unroll
    for (int k = 0; k < 32; ++k) acc += Usm[i * 34 + k] * uj[k];
    Fsm[i * 34 + lid] = acc;  // F[i][lid]
  }
  __syncthreads();

  // ================= Linear(1024 -> 4) + log_softmax =======================
  float p0 = 0.f, p1 = 0.f, p2 = 0.f, p3 = 0.f;
#pragma unroll 4
  for (int i = 0; i < 32; ++i) {
    float fv = Fsm[i * 34 + lid];
    int base = i * 32 + lid;  // row-major flatten F[i][j]
    p0 = fmaf(Wc[base], fv, p0);
    p1 = fmaf(Wc[1024 + base], fv, p1);
    p2 = fmaf(Wc[2048 + base], fv, p2);
    p3 = fmaf(Wc[3072 + base], fv, p3);
  }
  float pr[4] = {p0, p1, p2, p3};
#pragma unroll
  for (int o = 0; o < 4; ++o) {
    float v = pr[o];
    for (int off = 16; off > 0; off >>= 1) v += __shfl_down(v, off, 32);
    if (lid == 0) logit_s[o] = v + bc[o];
  }
  __syncthreads();
  if (lid < 4) {
    float l0 = logit_s[0], l1 = logit_s[1], l2 = logit_s[2], l3 = logit_s[3];
    float m = fmaxf(fmaxf(l0, l1), fmaxf(l2, l3));
    float lse = m + logf(expf(l0 - m) + expf(l1 - m) + expf(l2 - m) + expf(l3 - m));
    out[b * 4 + lid] = logit_s[lid] - lse;
  }
}

extern "C" void kernel_launch(void* const* d_in, const int* in_sizes, int n_in,
                              void* d_out, int out_size, void* d_ws, size_t ws_size,
                              hipStream_t stream) {
  const float* x  = (const float*)d_in[0];
  const float* W  = (const float*)d_in[1];
  const float* Wc = (const float*)d_in[2];
  const float* bc = (const float*)d_in[3];
  float* out = (float*)d_out;
  int B = in_sizes[0] / (C_ * T_);  // 4096
  spdnet_fused<<<B, 32, 0, stream>>>(x, W, Wc, bc, out);
  (void)n_in; (void)out_size; (void)d_ws; (void)ws_size;
}